// Transformer_55327768708466
// MI455X (gfx1250) — compile-verified
//
#include <hip/hip_runtime.h>
#include <hip/hip_bf16.h>
#include <math.h>

// ---------------------------------------------------------------------------
// Sizes from the reference
// ---------------------------------------------------------------------------
#define D_    196
#define NCH_  64
#define H_    8
#define DH_   64
#define INNER_ 512
#define MLP_  784
#define SEQ_  65

typedef float v2f_t __attribute__((ext_vector_type(2)));
typedef float v8f_t __attribute__((ext_vector_type(8)));

// ---------------------------------------------------------------------------
// Wave-level 16x16 output tile GEMM using V_WMMA_F32_16X16X4_F32.
// MASKED=false: caller guarantees the whole tile and all K rows are in
// bounds -> pure load+wmma loop.  MASKED=true: indices clamped (loads stay
// address-safe) and bounds applied as 0/1 value masks -> no divergent
// control flow either way; EXEC stays all-ones as WMMA requires.
// Fragment layouts per CDNA5 ISA 7.12.2:
//   A 16x4 : lanes 0-15 -> M=lane, v0=K0,v1=K1 ; lanes 16-31 -> v0=K2,v1=K3
//   B 4x16 : lanes 0-15 -> N=lane, v0=K0,v1=K1 ; lanes 16-31 -> v0=K2,v1=K3
//   C/D    : VGPR r: lanes 0-15 M=r, lanes 16-31 M=r+8, N=lane&15
// K must be a multiple of 4.  A must have >= K valid columns per row.
// ---------------------------------------------------------------------------
template <bool MASKED>
__device__ __forceinline__ void wave_gemm_nn(
    const float* __restrict__ A, int lda, int arows,
    const float* __restrict__ B, int ldb, int brows, int bcols,
    int K, int mbase, int nbase, v8f_t& acc)
{
  const int lane = threadIdx.x & 31;
  const int half = lane >> 4;
  const int l16  = lane & 15;
  const int ar   = mbase + l16;
  const int bc   = nbase + l16;
  float am = 1.0f, bm = 1.0f;
  int arc = ar, bcc = bc;
  if constexpr (MASKED) {
    am  = (ar < arows) ? 1.0f : 0.0f;
    bm  = (bc < bcols) ? 1.0f : 0.0f;
    arc = (ar < arows) ? ar : 0;   // clamp -> loads always address-safe
    bcc = (bc < bcols) ? bc : 0;
  }
  const float* __restrict__ Ap = A + (size_t)arc * lda;
  const float* __restrict__ Bp = B + bcc;
  for (int k = 0; k < K; k += 4) {
    const int kk = k + 2 * half;
    v2f_t a = *(const v2f_t*)(Ap + kk);      // contiguous, 8B aligned
    v2f_t b;
    if constexpr (MASKED) {
      const int   k0  = (kk     < brows) ? kk     : 0;
      const int   k1  = (kk + 1 < brows) ? kk + 1 : 0;
      const float bm0 = (kk     < brows) ? bm : 0.0f;
      const float bm1 = (kk + 1 < brows) ? bm : 0.0f;
      a.x *= am; a.y *= am;
      b.x = Bp[(size_t)k0 * ldb] * bm0;
      b.y = Bp[(size_t)k1 * ldb] * bm1;
    } else {
      b.x = Bp[(size_t)kk * ldb];
      b.y = Bp[(size_t)(kk + 1) * ldb];
    }
    acc = __builtin_amdgcn_wmma_f32_16x16x4_f32(false, a, false, b,
                                                (short)0, acc, false, false);
  }
}

// A (M x K) times B^T where Bt is N x K row-major (used for Q @ K^T).
template <bool MASKED>
__device__ __forceinline__ void wave_gemm_nt(
    const float* __restrict__ A, int lda, int arows,
    const float* __restrict__ Bt, int ldbt, int bnrows,
    int K, int mbase, int nbase, v8f_t& acc)
{
  const int lane = threadIdx.x & 31;
  const int half = lane >> 4;
  const int l16  = lane & 15;
  const int ar   = mbase + l16;
  const int bc   = nbase + l16;
  float am = 1.0f, bm = 1.0f;
  int arc = ar, bcc = bc;
  if constexpr (MASKED) {
    am  = (ar < arows) ? 1.0f : 0.0f;
    bm  = (bc < bnrows) ? 1.0f : 0.0f;
    arc = (ar < arows) ? ar : 0;
    bcc = (bc < bnrows) ? bc : 0;
  }
  const float* __restrict__ Ap = A + (size_t)arc * lda;
  const float* __restrict__ Bp = Bt + (size_t)bcc * ldbt;
  for (int k = 0; k < K; k += 4) {
    const int kk = k + 2 * half;
    v2f_t a = *(const v2f_t*)(Ap + kk);
    v2f_t b = *(const v2f_t*)(Bp + kk);
    if constexpr (MASKED) {
      a.x *= am; a.y *= am;
      b.x *= bm; b.y *= bm;
    }
    acc = __builtin_amdgcn_wmma_f32_16x16x4_f32(false, a, false, b,
                                                (short)0, acc, false, false);
  }
}

template <bool MASKED>
__device__ __forceinline__ void store_tile(
    float* __restrict__ Cp, int ldc, int crows, int ccols,
    int mbase, int nbase, const v8f_t& acc)
{
  const int lane = threadIdx.x & 31;
  const int half = lane >> 4;
  const int l16  = lane & 15;
  const int col  = nbase + l16;
  if constexpr (MASKED) {
    if (col < ccols) {
#pragma unroll
      for (int e = 0; e < 8; ++e) {
        const int row = mbase + e + 8 * half;
        if (row < crows) Cp[(size_t)row * ldc + col] = acc[e];
      }
    }
  } else {
#pragma unroll
    for (int e = 0; e < 8; ++e) {
      const int row = mbase + e + 8 * half;
      Cp[(size_t)row * ldc + col] = acc[e];
    }
  }
}

// ---------------------------------------------------------------------------
// K1: conv1d+BN+ReLU, sort-based channel gating, x_att build, LN1 -> h1
// ---------------------------------------------------------------------------
__global__ __launch_bounds__(256) void k_prep(
    const float* __restrict__ x, const float* __restrict__ tokens,
    const float* __restrict__ xpe, const float* __restrict__ convk,
    const float* __restrict__ bn_g, const float* __restrict__ bn_b,
    const float* __restrict__ bn_rm, const float* __restrict__ bn_rv,
    const float* __restrict__ fc_w1, const float* __restrict__ fc_w2,
    const float* __restrict__ ln1_g, const float* __restrict__ ln1_b,
    float* __restrict__ h1, float* __restrict__ xatt)
{
  __shared__ float as[NCH_ * D_];
  __shared__ float b1s[NCH_], b2s[NCH_], cs[NCH_], t1s[12];
  __shared__ float mus[SEQ_], rss[SEQ_];
  __shared__ int   midS;
  const int tid = threadIdx.x;

  const float km0 = convk[3], km1 = convk[4], km2 = convk[5]; // conv_k[1][:]
  const float bg = bn_g[0], bb = bn_b[0], rm = bn_rm[0];
  const float rstd = rsqrtf(bn_rv[0] + 1e-5f);

  for (int i = tid; i < NCH_ * D_; i += 256) {
    const int ch = i / D_, j = i - ch * D_;
    const float* r = xpe + ch * D_;
    const float xm = (j > 0)      ? r[j - 1] : 0.0f;
    const float xc = r[j];
    const float xp = (j < D_ - 1) ? r[j + 1] : 0.0f;
    const float conv = km0 * xm + km1 * xc + km2 * xp;
    const float bn = (conv - rm) * rstd * bg + bb;
    as[i] = bn > 0.0f ? bn : 0.0f;
  }
  __syncthreads();

  if (tid < NCH_) {
    float s = 0.0f;
    for (int j = 0; j < D_; ++j) s += as[tid * D_ + j];
    b1s[tid] = s * (1.0f / (float)D_);
  }
  __syncthreads();

  if (tid == 0) {
    float mx = -1e30f, mn = 1e30f; int nonpos = 0;
    for (int i = 0; i < NCH_; ++i) {
      const float v = b1s[i];
      mx = fmaxf(mx, v); mn = fminf(mn, v);
      if (v <= 0.0f) nonpos++;
    }
    int mid;
    if (mx < 0.0f || mn > 0.0f) mid = 32;
    else                        mid = (mx > 0.0f) ? nonpos : 0;
    midS = mid;
  }
  __syncthreads();

  if (tid < NCH_) {
    const float v = b1s[tid];
    int rank = 0;
    for (int j = 0; j < NCH_; ++j) {
      const float w = b1s[j];
      rank += (w < v) || (w == v && j < tid);
    }
    const int mid = midS;
    const float ls = (float)mid, le = (float)(NCH_ - mid);
    float out;
    if (rank < mid) out = v - 1.0f / (1.0f + powf(ls,  v));
    else            out = v + 1.0f / (1.0f + powf(le, -v));
    b2s[tid] = out;
  }
  __syncthreads();

  if (tid < 12) {
    float s = 0.0f;
    for (int ch = 0; ch < NCH_; ++ch) s += b2s[ch] * fc_w1[ch * 12 + tid];
    t1s[tid] = s > 0.0f ? s : 0.0f;
  }
  __syncthreads();

  if (tid < NCH_) {
    float s = 0.0f;
    for (int k = 0; k < 12; ++k) s += t1s[k] * fc_w2[k * NCH_ + tid];
    cs[tid] = 1.0f / (1.0f + expf(-s));
  }
  __syncthreads();

  for (int i = tid; i < SEQ_ * D_; i += 256) {
    const int row = i / D_, j = i - row * D_;
    xatt[i] = (row < NCH_) ? as[row * D_ + j] * cs[row] : tokens[j];
  }

  if (tid < SEQ_) {
    const float* r = x + tid * D_;
    float s = 0.0f;
    for (int j = 0; j < D_; ++j) s += r[j];
    const float m = s * (1.0f / (float)D_);
    float v = 0.0f;
    for (int j = 0; j < D_; ++j) { const float d = r[j] - m; v += d * d; }
    mus[tid] = m;
    rss[tid] = rsqrtf(v * (1.0f / (float)D_) + 1e-5f);
  }
  __syncthreads();
  for (int i = tid; i < SEQ_ * D_; i += 256) {
    const int row = i / D_, j = i - row * D_;
    h1[i] = (x[i] - mus[row]) * rss[row] * ln1_g[j] + ln1_b[j];
  }
}

// ---------------------------------------------------------------------------
// K2: qkv = h1[65,196] @ w_qkv[196,1536].  grid (12,5), one tile per wave.
// M tiles 0..3 are interior -> mask-free path (wave-uniform dispatch).
// ---------------------------------------------------------------------------
__global__ __launch_bounds__(256) void k_qkv(
    const float* __restrict__ h1, const float* __restrict__ wqkv,
    float* __restrict__ qkv)
{
  __builtin_prefetch(wqkv + (size_t)blockIdx.x * 128, 0, 3);
  const int wave  = threadIdx.x >> 5;
  const int mbase = blockIdx.y * 16;
  const int nt    = blockIdx.x * 8 + wave;      // 0..95
  v8f_t acc = {};
  if (mbase + 16 <= SEQ_) {
    wave_gemm_nn<false>(h1, D_, SEQ_, wqkv, 3 * INNER_, D_, 3 * INNER_,
                        D_, mbase, nt * 16, acc);
    store_tile<false>(qkv, 3 * INNER_, SEQ_, 3 * INNER_, mbase, nt * 16, acc);
  } else {
    wave_gemm_nn<true>(h1, D_, SEQ_, wqkv, 3 * INNER_, D_, 3 * INNER_,
                       D_, mbase, nt * 16, acc);
    store_tile<true>(qkv, 3 * INNER_, SEQ_, 3 * INNER_, mbase, nt * 16, acc);
  }
}

// ---------------------------------------------------------------------------
// K3: one block per head.  Q,K tiles + padded score matrix in LDS, V read
// from L2-resident global.  scores padded to [80][68]; softmax over 65 cols.
// ---------------------------------------------------------------------------
__global__ __launch_bounds__(256) void k_attn(
    const float* __restrict__ qkv, float* __restrict__ o)
{
  __shared__ alignas(16) float qs[SEQ_ * DH_];
  __shared__ alignas(16) float ksh[SEQ_ * DH_];
  __shared__ alignas(16) float ps[80 * 68];
  const int hd   = blockIdx.x;
  const int tid  = threadIdx.x;
  const int wave = tid >> 5;

  for (int i = tid; i < SEQ_ * DH_; i += 256) {
    const int row = i >> 6, d = i & 63;
    qs[i]  = qkv[row * (3 * INNER_) + hd * DH_ + d];
    ksh[i] = qkv[row * (3 * INNER_) + INNER_ + hd * DH_ + d];
  }
  __syncthreads();

  // scores = (Q @ K^T) * DH^-0.5; tiles 5x5 over padded 80x80; stores clipped
  // to the 80x68 LDS array (cols>=65 are exactly 0 from zero-masked K rows).
  for (int job = wave; job < 25; job += 8) {
    const int mt = job / 5, nt = job % 5;
    v8f_t acc = {};
    if (mt < 4 && nt < 4)
      wave_gemm_nt<false>(qs, DH_, SEQ_, ksh, DH_, SEQ_, DH_,
                          mt * 16, nt * 16, acc);
    else
      wave_gemm_nt<true>(qs, DH_, SEQ_, ksh, DH_, SEQ_, DH_,
                         mt * 16, nt * 16, acc);
#pragma unroll
    for (int e = 0; e < 8; ++e) acc[e] *= 0.125f;   // 64^-0.5
    if (nt < 4)  // rows always < 80; only col tile 4 needs the 68-col clip
      store_tile<false>(ps, 68, 80, 68, mt * 16, nt * 16, acc);
    else
      store_tile<true>(ps, 68, 80, 68, mt * 16, nt * 16, acc);
  }
  __syncthreads();

  for (int row = tid; row < SEQ_; row += 256) {
    float* pr = ps + row * 68;
    float mx = -1e30f;
    for (int j = 0; j < SEQ_; ++j) mx = fmaxf(mx, pr[j]);
    float s = 0.0f;
    for (int j = 0; j < SEQ_; ++j) { const float e = expf(pr[j] - mx); pr[j] = e; s += e; }
    const float inv = 1.0f / s;
    for (int j = 0; j < SEQ_; ++j) pr[j] *= inv;
    pr[65] = pr[66] = pr[67] = 0.0f;   // K padded to 68 for the P@V GEMM
  }
  __syncthreads();

  // O_head = P[65,68pad] @ V.  Mask-free K=64 main loop (V rows 0..63 valid),
  // then one masked K=4 tail step covering absolute rows 64..67 (only 64 real).
  const float* __restrict__ Vb = qkv + 2 * INNER_ + hd * DH_;
  for (int job = wave; job < 20; job += 8) {
    const int mt = job / 4, nt = job % 4;
    v8f_t acc = {};
    wave_gemm_nn<false>(ps, 68, 80, Vb, 3 * INNER_, 64, DH_,
                        64, mt * 16, nt * 16, acc);
    // tail: A cols 64..67 of ps, B rows 64..67 (relative brows=1 -> row 64)
    wave_gemm_nn<true>(ps + 64, 68, 80, Vb + (size_t)64 * (3 * INNER_),
                       3 * INNER_, 1, DH_, 4, mt * 16, nt * 16, acc);
    if (mt < 4)
      store_tile<false>(o + hd * DH_, INNER_, SEQ_, DH_, mt * 16, nt * 16, acc);
    else
      store_tile<true>(o + hd * DH_, INNER_, SEQ_, DH_, mt * 16, nt * 16, acc);
  }
}

// ---------------------------------------------------------------------------
// K4: x2 = o @ w_out + b_out + x + x_att ; h2 = LN2(x2).  One block per
// 16-row stripe so the LN can run on the freshly built rows in LDS.
// ---------------------------------------------------------------------------
__global__ __launch_bounds__(256) void k_out(
    const float* __restrict__ o, const float* __restrict__ wout,
    const float* __restrict__ bout, const float* __restrict__ x,
    const float* __restrict__ xatt,
    const float* __restrict__ ln2_g, const float* __restrict__ ln2_b,
    float* __restrict__ x2, float* __restrict__ h2)
{
  __shared__ alignas(16) float xt[16 * D_];
  __shared__ float mus[16], rss[16];
  const int tid  = threadIdx.x;
  const int wave = tid >> 5;
  const int lane = tid & 31;
  const int half = lane >> 4, l16 = lane & 15;
  const int mbase = blockIdx.x * 16;
  const bool mok = (mbase + 16 <= SEQ_);

  __builtin_prefetch(wout, 0, 3);
  for (int nt = wave; nt < 13; nt += 8) {
    v8f_t acc = {};
    const int col = nt * 16 + l16;
    if (mok && nt < 12) {
      wave_gemm_nn<false>(o, INNER_, SEQ_, wout, D_, INNER_, D_,
                          INNER_, mbase, nt * 16, acc);
#pragma unroll
      for (int e = 0; e < 8; ++e) {
        const int row = mbase + e + 8 * half;
        const float v = acc[e] + bout[col] + x[row * D_ + col] + xatt[row * D_ + col];
        xt[(row - mbase) * D_ + col] = v;
        x2[row * D_ + col] = v;
      }
    } else {
      wave_gemm_nn<true>(o, INNER_, SEQ_, wout, D_, INNER_, D_,
                         INNER_, mbase, nt * 16, acc);
      if (col < D_) {
#pragma unroll
        for (int e = 0; e < 8; ++e) {
          const int row = mbase + e + 8 * half;
          if (row < SEQ_) {
            const float v = acc[e] + bout[col] + x[row * D_ + col] + xatt[row * D_ + col];
            xt[(row - mbase) * D_ + col] = v;
            x2[row * D_ + col] = v;
          }
        }
      }
    }
  }
  __syncthreads();

  if (tid < 16 && mbase + tid < SEQ_) {
    const float* r = xt + tid * D_;
    float s = 0.0f;
    for (int j = 0; j < D_; ++j) s += r[j];
    const float m = s * (1.0f / (float)D_);
    float v = 0.0f;
    for (int j = 0; j < D_; ++j) { const float d = r[j] - m; v += d * d; }
    mus[tid] = m;
    rss[tid] = rsqrtf(v * (1.0f / (float)D_) + 1e-5f);
  }
  __syncthreads();

  for (int i = tid; i < 16 * D_; i += 256) {
    const int r = i / D_, j = i - r * D_;
    const int row = mbase + r;
    if (row < SEQ_)
      h2[row * D_ + j] = (xt[i] - mus[r]) * rss[r] * ln2_g[j] + ln2_b[j];
  }
}

// ---------------------------------------------------------------------------
// K5: g = gelu(h2 @ ff_w1 + ff_b1)   (exact gelu, erf form)
// ---------------------------------------------------------------------------
__global__ __launch_bounds__(256) void k_ff1(
    const float* __restrict__ h2, const float* __restrict__ w1,
    const float* __restrict__ b1, float* __restrict__ g)
{
  const int wave = threadIdx.x >> 5;
  const int lane = threadIdx.x & 31;
  const int half = lane >> 4, l16 = lane & 15;
  const int mbase = blockIdx.y * 16;
  const int nt = blockIdx.x * 8 + wave;
  if (nt >= 49) return;                       // wave-uniform exit
  __builtin_prefetch(w1 + (size_t)nt * 16, 0, 3);
  v8f_t acc = {};
  const int col = nt * 16 + l16;              // always < 784
  if (mbase + 16 <= SEQ_) {
    wave_gemm_nn<false>(h2, D_, SEQ_, w1, MLP_, D_, MLP_, D_,
                        mbase, nt * 16, acc);
#pragma unroll
    for (int e = 0; e < 8; ++e) {
      const int row = mbase + e + 8 * half;
      const float v = acc[e] + b1[col];
      g[row * MLP_ + col] = 0.5f * v * (1.0f + erff(v * 0.70710678118654752f));
    }
  } else {
    wave_gemm_nn<true>(h2, D_, SEQ_, w1, MLP_, D_, MLP_, D_,
                       mbase, nt * 16, acc);
#pragma unroll
    for (int e = 0; e < 8; ++e) {
      const int row = mbase + e + 8 * half;
      if (row < SEQ_) {
        const float v = acc[e] + b1[col];
        g[row * MLP_ + col] = 0.5f * v * (1.0f + erff(v * 0.70710678118654752f));
      }
    }
  }
}

// ---------------------------------------------------------------------------
// K6: out = g @ ff_w2 + ff_b2 + x2
// ---------------------------------------------------------------------------
__global__ __launch_bounds__(256) void k_ff2(
    const float* __restrict__ g, const float* __restrict__ w2,
    const float* __restrict__ b2, const float* __restrict__ x2,
    float* __restrict__ out)
{
  const int wave = threadIdx.x >> 5;
  const int lane = threadIdx.x & 31;
  const int half = lane >> 4, l16 = lane & 15;
  const int mbase = blockIdx.y * 16;
  const int nt = blockIdx.x * 8 + wave;
  if (nt >= 13) return;                       // wave-uniform exit
  v8f_t acc = {};
  const int col = nt * 16 + l16;
  if (mbase + 16 <= SEQ_ && nt < 12) {
    wave_gemm_nn<false>(g, MLP_, SEQ_, w2, D_, MLP_, D_, MLP_,
                        mbase, nt * 16, acc);
#pragma unroll
    for (int e = 0; e < 8; ++e) {
      const int row = mbase + e + 8 * half;
      out[row * D_ + col] = acc[e] + b2[col] + x2[row * D_ + col];
    }
  } else {
    wave_gemm_nn<true>(g, MLP_, SEQ_, w2, D_, MLP_, D_, MLP_,
                       mbase, nt * 16, acc);
    if (col < D_) {
#pragma unroll
      for (int e = 0; e < 8; ++e) {
        const int row = mbase + e + 8 * half;
        if (row < SEQ_)
          out[row * D_ + col] = acc[e] + b2[col] + x2[row * D_ + col];
      }
    }
  }
}

// ---------------------------------------------------------------------------
// Launch
// ---------------------------------------------------------------------------
extern "C" void kernel_launch(void* const* d_in, const int* in_sizes, int n_in,
                              void* d_out, int out_size, void* d_ws, size_t ws_size,
                              hipStream_t stream) {
  const float* x      = (const float*)d_in[0];
  const float* tokens = (const float*)d_in[1];
  const float* xpe    = (const float*)d_in[2];
  const float* convk  = (const float*)d_in[3];
  const float* bn_g   = (const float*)d_in[4];
  const float* bn_b   = (const float*)d_in[5];
  const float* bn_rm  = (const float*)d_in[6];
  const float* bn_rv  = (const float*)d_in[7];
  const float* fc_w1  = (const float*)d_in[8];
  const float* fc_w2  = (const float*)d_in[9];
  const float* ln1_g  = (const float*)d_in[10];
  const float* ln1_b  = (const float*)d_in[11];
  const float* ln2_g  = (const float*)d_in[12];
  const float* ln2_b  = (const float*)d_in[13];
  const float* w_qkv  = (const float*)d_in[14];
  const float* w_out  = (const float*)d_in[15];
  const float* b_out  = (const float*)d_in[16];
  const float* ff_w1  = (const float*)d_in[17];
  const float* ff_b1  = (const float*)d_in[18];
  const float* ff_w2  = (const float*)d_in[19];
  const float* ff_b2  = (const float*)d_in[20];

  float* ws   = (float*)d_ws;
  float* h1   = ws;                       // 65*196   = 12740
  float* xatt = ws + 12740;               // 65*196   = 12740
  float* qkv  = ws + 25480;               // 65*1536  = 99840
  float* o    = ws + 125320;              // 65*512   = 33280
  float* x2   = ws + 158600;              // 65*196   = 12740
  float* h2   = ws + 171340;              // 65*196   = 12740
  float* gbuf = ws + 184080;              // 65*784   = 50960  (end 235040 floats)

  k_prep<<<dim3(1), dim3(256), 0, stream>>>(x, tokens, xpe, convk,
                                            bn_g, bn_b, bn_rm, bn_rv,
                                            fc_w1, fc_w2, ln1_g, ln1_b,
                                            h1, xatt);
  k_qkv <<<dim3(12, 5), dim3(256), 0, stream>>>(h1, w_qkv, qkv);
  k_attn<<<dim3(8),     dim3(256), 0, stream>>>(qkv, o);
  k_out <<<dim3(5),     dim3(256), 0, stream>>>(o, w_out, b_out, x, xatt,
                                                ln2_g, ln2_b, x2, h2);
  k_ff1 <<<dim3(7, 5),  dim3(256), 0, stream>>>(h2, ff_w1, ff_b1, gbuf);
  k_ff2 <<<dim3(2, 5),  dim3(256), 0, stream>>>(gbuf, ff_w2, ff_b2, x2,
                                                (float*)d_out);
}